// RBF_30150670418211
// MI455X (gfx1250) — compile-verified
//
#include <hip/hip_runtime.h>

// RBF kernel (SVGD median-heuristic) for MI455X / gfx1250.
// Memory-bound: 270 MB of outputs vs 67 MFLOP of math -> single-write fusion,
// inputs resident in LDS, fp32 WMMA for the Gram matrix, exact radix-select
// median, non-temporal streaming stores for the write-once outputs.

typedef __attribute__((ext_vector_type(2))) float v2f;
typedef __attribute__((ext_vector_type(4))) float v4f;
typedef __attribute__((ext_vector_type(8))) float v8f;

#define NB 128
#define NN 64
#define ND 64
#define INV_LN65 0.23955593f   // 1 / ln(N+1) = 1 / ln(65)

// ---------------------------------------------------------------------------
// Kernel 1: per-batch dist_sq via WMMA, exact lower median, gamma, kappa.
// One block per batch; 256 threads = 8 wave32.
// ---------------------------------------------------------------------------
__global__ __launch_bounds__(256) void rbf_phase1(
    const float* __restrict__ X, const float* __restrict__ Y,
    float* __restrict__ kappa_out, float* __restrict__ gamma_out)
{
    __shared__ float xs[NN * ND];      // 16 KB
    __shared__ float ys[NN * ND];      // 16 KB
    __shared__ float dist[NN * NN];    // 16 KB
    __shared__ float xn[NN];
    __shared__ float yn[NN];
    __shared__ int   s_cnt;

    const int b   = blockIdx.x;
    const int tid = threadIdx.x;

    // ---- stage A: inputs -> LDS (128-bit wide, coalesced) ----
    {
        const v4f* x4 = (const v4f*)(X + (size_t)b * NN * ND);
        const v4f* y4 = (const v4f*)(Y + (size_t)b * NN * ND);
        v4f* xs4 = (v4f*)xs;
        v4f* ys4 = (v4f*)ys;
        for (int i = tid; i < NN * ND / 4; i += 256) {
            xs4[i] = x4[i];
            ys4[i] = y4[i];
        }
    }
    __syncthreads();

    // ---- stage B: row norms ||x_i||^2, ||y_j||^2 ----
    if (tid < 2 * NN) {
        const float* src = (tid < NN) ? (xs + tid * ND) : (ys + (tid - NN) * ND);
        float s = 0.f;
        #pragma unroll 8
        for (int d = 0; d < ND; ++d) { float v = src[d]; s += v * v; }
        if (tid < NN) xn[tid] = s; else yn[tid - NN] = s;
    }
    __syncthreads();

    // ---- stage C: G = X * Y^T with V_WMMA_F32_16X16X4_F32;
    //               dist_sq = xn + yn - 2G.
    // 4x4 grid of 16x16 output tiles; each of the 8 waves owns 2 tiles.
    {
        const int wave = tid >> 5;        // 0..7 (uniform per wave)
        const int lane = tid & 31;
        const int m    = lane & 15;
        const int hi   = lane >> 4;       // which lane-half
        #pragma unroll
        for (int t = wave * 2; t < wave * 2 + 2; ++t) {
            const int i0 = (t >> 2) * 16;
            const int j0 = (t & 3) * 16;
            v8f acc = {};
            #pragma unroll
            for (int kk = 0; kk < ND / 4; ++kk) {
                // A 16x4 f32: VGPR0 = {K=k0 (lanes 0-15), K=k0+2 (lanes 16-31)},
                //             VGPR1 = {K=k0+1, K=k0+3}  -> contiguous b64 LDS read.
                const int k0 = kk * 4 + 2 * hi;
                v2f a, bf;
                a.x  = xs[(i0 + m) * ND + k0];
                a.y  = xs[(i0 + m) * ND + k0 + 1];
                bf.x = ys[(j0 + m) * ND + k0];   // B = Y^T, row K, col n -> y[j0+n][k]
                bf.y = ys[(j0 + m) * ND + k0 + 1];
                acc = __builtin_amdgcn_wmma_f32_16x16x4_f32(
                    false, a, false, bf, (short)0, acc, false, false);
            }
            #pragma unroll
            for (int r = 0; r < 8; ++r) {
                const int i = i0 + r + 8 * hi;   // C/D layout: M = r + 8*hi
                const int j = j0 + m;            //             N = lane&15
                dist[i * NN + j] = xn[i] + yn[j] - 2.0f * acc[r];
            }
        }
    }
    __syncthreads();

    // ---- stage D: exact lower median = rank-2047 of 4096 values.
    // Radix-select on raw IEEE bits (monotone for non-negative floats).
    unsigned prefix = 0;
    int want = (NN * NN - 1) / 2;   // 2047
    for (int bit = 31; bit >= 0; --bit) {
        const unsigned bm = 1u << bit;
        const unsigned hm = ~((bm << 1) - 1u);   // bits strictly above 'bit' (0 when bit==31)
        if (tid == 0) s_cnt = 0;
        __syncthreads();
        int c = 0;
        for (int idx = tid; idx < NN * NN; idx += 256) {
            const unsigned u = __float_as_uint(dist[idx]);
            if ((u & hm) == prefix && (u & bm) == 0u) ++c;
        }
        atomicAdd(&s_cnt, c);
        __syncthreads();
        const int cnt = s_cnt;                   // every thread follows same path
        if (want >= cnt) { want -= cnt; prefix |= bm; }
        __syncthreads();                         // protect s_cnt reuse
    }
    const float med   = __uint_as_float(prefix);
    // h = med/(2 ln65); gamma = 1/(1e-8 + 2 sigma^2) = 1/(1e-8 + med/ln65)
    const float gamma = 1.0f / (1e-8f + med * INV_LN65);
    if (tid == 0) gamma_out[b] = gamma;

    // ---- stage E: kappa = exp(-gamma * dist_sq) (2 MB; stays in L2, re-read
    //               by kernel 2 -> keep regular-temporal stores) ----
    for (int idx = tid; idx < NN * NN; idx += 256)
        kappa_out[(size_t)b * NN * NN + idx] = __expf(-gamma * dist[idx]);
}

// ---------------------------------------------------------------------------
// Kernel 2: stream diff and kappa_grad (2 x 134 MB) with non-temporal 128-bit
// stores (write-once, never re-read -> don't thrash the 192 MB L2).
// Grid = 128 batches x 8 row-chunks = 1024 blocks for HBM saturation.
// diff is recomputed from LDS-resident inputs -> never round-trips HBM.
// ---------------------------------------------------------------------------
__global__ __launch_bounds__(256) void rbf_phase2(
    const float* __restrict__ X, const float* __restrict__ Y,
    const float* __restrict__ kappa, const float* __restrict__ gamma_in,
    float* __restrict__ diff_out, float* __restrict__ kgrad_out)
{
    __shared__ v4f  ys4[NN * ND / 4];   // 16 KB: all of y for this batch
    __shared__ v4f  xs4[8 * ND / 4];    //  2 KB: 8 rows of x
    __shared__ float kap[8 * NN];       //  2 KB: matching kappa rows

    const int tid = threadIdx.x;
    const int bid = blockIdx.x;
    const int b   = bid >> 3;
    const int i0  = (bid & 7) * 8;

    const v4f* y4 = (const v4f*)(Y + (size_t)b * NN * ND);
    const v4f* x4 = (const v4f*)(X + (size_t)b * NN * ND + (size_t)i0 * ND);

    for (int i = tid; i < NN * ND / 4; i += 256) ys4[i] = y4[i];
    if (tid < 8 * ND / 4) xs4[tid] = x4[tid];
    for (int i = tid; i < 8 * NN; i += 256)
        kap[i] = kappa[(size_t)b * NN * NN + (size_t)i0 * NN + i];
    const float gamma = gamma_in[b];      // uniform -> scalar load
    __syncthreads();

    // Block covers a contiguous [b, i0..i0+7] slab of both outputs.
    v4f* dptr = (v4f*)diff_out  + (size_t)bid * (8 * NN * ND / 4);
    v4f* gptr = (v4f*)kgrad_out + (size_t)bid * (8 * NN * ND / 4);

    // idx = ii*1024 + j*16 + d4 ; consecutive lanes -> consecutive 16B stores.
    for (int idx = tid; idx < 8 * NN * ND / 4; idx += 256) {
        const int d4 = idx & (ND / 4 - 1);
        const int j  = (idx >> 4) & (NN - 1);
        const int ii = idx >> 10;
        const v4f xv = xs4[ii * (ND / 4) + d4];
        const v4f yv = ys4[j * (ND / 4) + d4];
        const v4f dx = xv - yv;
        const float s = -2.0f * gamma * kap[ii * NN + j];
        const v4f kg = dx * s;
        __builtin_nontemporal_store(dx, &dptr[idx]);   // TH = NT streaming store
        __builtin_nontemporal_store(kg, &gptr[idx]);   // TH = NT streaming store
    }
}

// ---------------------------------------------------------------------------
extern "C" void kernel_launch(void* const* d_in, const int* in_sizes, int n_in,
                              void* d_out, int out_size, void* d_ws, size_t ws_size,
                              hipStream_t stream)
{
    const float* X = (const float*)d_in[0];   // [128,64,64]
    const float* Y = (const float*)d_in[1];   // [128,64,64]

    float* out   = (float*)d_out;
    float* kappa = out;                                    // [128,64,64]    (524288)
    float* diff  = kappa + (size_t)NB * NN * NN;           // [128,64,64,64] (33554432)
    float* gamma = diff  + (size_t)NB * NN * NN * ND;      // [128]
    float* kgrad = gamma + NB;                             // [128,64,64,64]

    rbf_phase1<<<NB, 256, 0, stream>>>(X, Y, kappa, gamma);
    rbf_phase2<<<NB * 8, 256, 0, stream>>>(X, Y, kappa, gamma, diff, kgrad);
}